// Winograd_Conv2d_6854767804818
// MI455X (gfx1250) — compile-verified
//
#include <hip/hip_runtime.h>

// CDNA5 (gfx1250) fused Winograd F(2x2,3x3) conv, fp32 via V_WMMA_F32_16X16X4_F32.
// Shapes fixed by the reference: N=16, C=32, K=32, H=W=128 -> out 126x126, T=63 tiles/dim.
//
// Operand layouts are pair-interleaved along the channel (WMMA-K) dimension so
// every A/B fragment is a single 64-bit load (global_load_b64 / ds_load_b64)
// instead of two b32 loads.

typedef __attribute__((ext_vector_type(2))) float v2f;
typedef __attribute__((ext_vector_type(8))) float v8f;

#define NN   16
#define CC   32
#define KK   32
#define HH   128
#define WW   128
#define TT   63    // tiles per spatial dim
#define OUTD 126   // output spatial dim

// ---------------------------------------------------------------------------
// Kernel 1: filter transform U = G g G^T, written to workspace as
// U[ij][c/2][k][2]  (channel pairs interleaved) so the main kernel's WMMA
// A-fragments are single b64 loads, 128B-contiguous per half-wave.
// 32x32 = 1024 threads, one (k,c) pair each.
// ---------------------------------------------------------------------------
__global__ __launch_bounds__(1024) void wg_filter_transform(
    const float* __restrict__ filt, float* __restrict__ U) {
  const int tid = threadIdx.x;          // 0..1023
  const int c = tid & 31;
  const int k = tid >> 5;
  const float* g = filt + (size_t)(k * CC + c) * 9;
  const float g0 = g[0], g1 = g[1], g2 = g[2];
  const float g3 = g[3], g4 = g[4], g5 = g[5];
  const float g6 = g[6], g7 = g[7], g8 = g[8];
  // t = G g   (G = [[1,0,0],[.5,.5,.5],[.5,-.5,.5],[0,0,1]])
  float t[4][3];
  t[0][0] = g0;                 t[0][1] = g1;                 t[0][2] = g2;
  t[1][0] = 0.5f*(g0+g3+g6);    t[1][1] = 0.5f*(g1+g4+g7);    t[1][2] = 0.5f*(g2+g5+g8);
  t[2][0] = 0.5f*(g0-g3+g6);    t[2][1] = 0.5f*(g1-g4+g7);    t[2][2] = 0.5f*(g2-g5+g8);
  t[3][0] = g6;                 t[3][1] = g7;                 t[3][2] = g8;
  // u = t G^T ; store pair-interleaved: idx = ((ij*16 + c/2)*KK + k)*2 + (c&1)
  const int pr = c >> 1;
  const int lo = c & 1;
#pragma unroll
  for (int i = 0; i < 4; ++i) {
    const float u0 = t[i][0];
    const float u1 = 0.5f * (t[i][0] + t[i][1] + t[i][2]);
    const float u2 = 0.5f * (t[i][0] - t[i][1] + t[i][2]);
    const float u3 = t[i][2];
    U[((((i*4 + 0) * 16 + pr) * KK) + k) * 2 + lo] = u0;
    U[((((i*4 + 1) * 16 + pr) * KK) + k) * 2 + lo] = u1;
    U[((((i*4 + 2) * 16 + pr) * KK) + k) * 2 + lo] = u2;
    U[((((i*4 + 3) * 16 + pr) * KK) + k) * 2 + lo] = u3;
  }
}

// ---------------------------------------------------------------------------
// Kernel 2: fused main pass. One workgroup = 256 threads (8 wave32) handles
// (n, tile-row t, 16 consecutive tile-cols u0..u0+15).
//   Phase 1: global -> regs, V = B^T d B -> LDS  Vs[ij][c/2][p][2]
//   Phase 2: per (ij, k-block): 8 x wmma_f32_16x16x4_f32 over C=32 -> Ms
//   Phase 3: Y = A^T M A -> global (predicated for ragged last block)
// ---------------------------------------------------------------------------
__global__ __launch_bounds__(256) void wg_main(
    const float* __restrict__ in, const float* __restrict__ U,
    float* __restrict__ out) {
  __shared__ float Vs[16 * CC * 16];   // 32 KB: [ij][c/2][p][2]
  __shared__ float Ms[16 * KK * 16];   // 32 KB: [ij][k][p]

  const int tid  = threadIdx.x;
  const int lane = tid & 31;
  const int wave = tid >> 5;
  const int n  = blockIdx.z;
  const int t  = blockIdx.y;
  const int u0 = blockIdx.x << 4;

  // ---- Phase 1: load 4x4 tiles (stride-2 overlap) + B^T d B --------------
#pragma unroll
  for (int it = 0; it < 2; ++it) {
    const int idx = tid + it * 256;    // 0..511 -> (c, p)
    const int p = idx & 15;
    const int c = idx >> 4;            // 0..31
    int u = u0 + p; if (u > TT - 1) u = TT - 1;   // clamp: keep EXEC full
    const float* src = in + ((size_t)(n * CC + c) * HH + 2 * t) * WW + 2 * u;
    float d[4][4];
#pragma unroll
    for (int i = 0; i < 4; ++i) {
      const float2 lo2 = *(const float2*)(src + (size_t)i * WW);      // 8B aligned
      const float2 hi2 = *(const float2*)(src + (size_t)i * WW + 2);
      d[i][0] = lo2.x; d[i][1] = lo2.y; d[i][2] = hi2.x; d[i][3] = hi2.y;
    }
    // rows: r = B^T d   (B^T rows: [1,0,-1,0],[0,1,1,0],[0,-1,1,0],[0,1,0,-1])
    float r[4][4];
#pragma unroll
    for (int j = 0; j < 4; ++j) {
      r[0][j] = d[0][j] - d[2][j];
      r[1][j] = d[1][j] + d[2][j];
      r[2][j] = d[2][j] - d[1][j];
      r[3][j] = d[1][j] - d[3][j];
    }
    // cols: V = r B ; store pair-interleaved for b64 B-fragment loads
    const int pr = c >> 1;
    const int lo = c & 1;
#pragma unroll
    for (int i = 0; i < 4; ++i) {
      const float v0 = r[i][0] - r[i][2];
      const float v1 = r[i][1] + r[i][2];
      const float v2 = r[i][2] - r[i][1];
      const float v3 = r[i][1] - r[i][3];
      Vs[(((i*4 + 0) * 16 + pr) * 16 + p) * 2 + lo] = v0;
      Vs[(((i*4 + 1) * 16 + pr) * 16 + p) * 2 + lo] = v1;
      Vs[(((i*4 + 2) * 16 + pr) * 16 + p) * 2 + lo] = v2;
      Vs[(((i*4 + 3) * 16 + pr) * 16 + p) * 2 + lo] = v3;
    }
  }
  __syncthreads();

  // ---- Phase 2: channel-contraction GEMMs via WMMA f32 16x16x4 -----------
  // 32 output tiles (16 ij-positions x 2 k-blocks); wave w owns 4 of them.
  // f32 16x16x4 fragment layouts (ISA 7.12.2):
  //   A (16x4): lanes 0-15 M=lane / lanes 16-31 M=lane-16; VGPR0={K0|K2}, VGPR1={K1|K3}
  //   B (4x16): lanes hold N; VGPR0={K0|K2}, VGPR1={K1|K3}
  //   C/D (16x16): VGPR v: lanes 0-15 -> M=v, lanes 16-31 -> M=v+8; N=lane&15
  // With pair-interleaving, each fragment (VGPR0,VGPR1) is one 64-bit load:
  // chunk cc -> lanes 0-15 read pair row 2cc, lanes 16-31 read pair row 2cc+1.
  const int hp   = (lane < 16) ? 0 : 1;     // which channel-pair this half-wave reads
  const int prow = lane & 15;
#pragma unroll
  for (int q = 0; q < 4; ++q) {
    const int ot = wave * 4 + q;       // 0..31
    const int ij = ot >> 1;
    const int kb = ot & 1;
    const int mrow = kb * 16 + prow;
    const float2* ub = (const float2*)U + (size_t)ij * (16 * KK);  // [pair][k]
    const float2* vb = (const float2*)Vs + ij * (16 * 16);         // [pair][p]
    v8f acc = {};
#pragma unroll
    for (int cc = 0; cc < 8; ++cc) {   // 8 chunks of 4 channels
      const int pr0 = cc * 2 + hp;
      const float2 av = ub[pr0 * KK + mrow];   // global_load_b64, L2-resident
      const float2 bv = vb[pr0 * 16 + prow];   // ds_load_b64, conflict-free
      v2f a, b;
      a.x = av.x; a.y = av.y;
      b.x = bv.x; b.y = bv.y;
      acc = __builtin_amdgcn_wmma_f32_16x16x4_f32(
          /*neg_a=*/false, a, /*neg_b=*/false, b,
          /*c_mod=*/(short)0, acc, /*reuse_a=*/false, /*reuse_b=*/false);
    }
    // D -> Ms[ij][k][p]
    const int koff = kb * 16 + ((lane < 16) ? 0 : 8);
    float* mb = Ms + ij * (KK * 16);
#pragma unroll
    for (int v = 0; v < 8; ++v)
      mb[(koff + v) * 16 + prow] = acc[v];
  }
  __syncthreads();

  // ---- Phase 3: inverse transform Y = A^T M A + store --------------------
#pragma unroll
  for (int it = 0; it < 2; ++it) {
    const int idx = tid + it * 256;    // (k, p)
    const int p = idx & 15;
    const int k = idx >> 4;
    const int u = u0 + p;
    if (u >= TT) continue;             // ragged last block
    float m[4][4];
#pragma unroll
    for (int i = 0; i < 4; ++i)
#pragma unroll
      for (int j = 0; j < 4; ++j)
        m[i][j] = Ms[((i*4 + j) * KK + k) * 16 + p];
    // A^T = [[1,1,1,0],[0,1,-1,-1]]
    float r0[4], r1[4];
#pragma unroll
    for (int j = 0; j < 4; ++j) {
      r0[j] = m[0][j] + m[1][j] + m[2][j];
      r1[j] = m[1][j] - m[2][j] - m[3][j];
    }
    const float y00 = r0[0] + r0[1] + r0[2];
    const float y01 = r0[1] - r0[2] - r0[3];
    const float y10 = r1[0] + r1[1] + r1[2];
    const float y11 = r1[1] - r1[2] - r1[3];
    float* dst = out + ((size_t)(n * KK + k) * OUTD + 2 * t) * OUTD + 2 * u;
    *(float2*)(dst)        = make_float2(y00, y01);   // 8B aligned (even offsets)
    *(float2*)(dst + OUTD) = make_float2(y10, y11);
  }
}

// ---------------------------------------------------------------------------
extern "C" void kernel_launch(void* const* d_in, const int* in_sizes, int n_in,
                              void* d_out, int out_size, void* d_ws, size_t ws_size,
                              hipStream_t stream) {
  (void)in_sizes; (void)n_in; (void)out_size; (void)ws_size;
  const float* in   = (const float*)d_in[0];   // (16,32,128,128) fp32
  const float* filt = (const float*)d_in[1];   // (32,32,3,3) fp32
  float* out = (float*)d_out;                  // (16,32,126,126) fp32
  float* U   = (float*)d_ws;                   // 16*32*32 floats = 64 KB

  wg_filter_transform<<<1, 1024, 0, stream>>>(filt, U);
  // grid: (ceil(63/16)=4 tile-col blocks, 63 tile rows, 16 images)
  wg_main<<<dim3(4, 63, 16), 256, 0, stream>>>(in, U, out);
}